// AMPS_FAST_22333829939476
// MI455X (gfx1250) — compile-verified
//
#include <hip/hip_runtime.h>
#include <cstdint>
#include <cstddef>

#ifndef __has_builtin
#define __has_builtin(x) 0
#endif

typedef float f4  __attribute__((ext_vector_type(4)));
typedef float v2f __attribute__((ext_vector_type(2)));
typedef float v8f __attribute__((ext_vector_type(8)));
typedef int   v4i __attribute__((ext_vector_type(4)));

#define NPOS    1024    // positions n = 0..1023
#define NCOL    1023    // chain columns 0..1022
#define SEG     64      // blocks per phase-1 segment
#define NSEGMAX 16      // max segments per column
#define CHUNK   8       // blocks per async chunk (1 KB per slot)
#define NCHUNKS 8       // SEG / CHUNK

// LDS layout per wave: [buf(2)][slot(4)] ; slot stride 1040B (16B aligned,
// 4-bank stagger so the 4 slots' broadcast reads don't conflict).
#define SLOT_STRIDE 1040
#define BUF_STRIDE  (4 * SLOT_STRIDE)       // 4160
#define WAVE_LDS    8448                     // 2*BUF_STRIDE rounded up

typedef __attribute__((address_space(1))) v4i gv4i_t;   // global int4
typedef __attribute__((address_space(3))) v4i lv4i_t;   // LDS int4

#if __has_builtin(__builtin_amdgcn_global_load_async_to_lds_b128)
__device__ __forceinline__ void async_b128(const char* g, char* l) {
  __builtin_amdgcn_global_load_async_to_lds_b128(
      (gv4i_t*)(uintptr_t)g, (lv4i_t*)(uint32_t)(uintptr_t)l, 0, 0);
}
#else
__device__ __forceinline__ void async_b128(const char* g, char* l) {
  unsigned loff = (unsigned)(uintptr_t)l;   // low 32 bits = LDS offset
  asm volatile("global_load_async_to_lds_b128 %0, %1, off"
               :: "v"(loff), "v"(g) : "memory");
}
#endif

#if __has_builtin(__builtin_amdgcn_s_wait_asynccnt)
#define WAIT_ASYNC(n) do { asm volatile("" ::: "memory");                    \
  __builtin_amdgcn_s_wait_asynccnt(n); asm volatile("" ::: "memory"); } while (0)
#else
#define WAIT_ASYNC(n) asm volatile("s_wait_asynccnt %0" :: "i"(n) : "memory")
#endif

// ---------------------------------------------------------------------------
// Phase 1: one block per column; wave k owns segments 4k..4k+3.
// Lane = slot(=seg within wave, bits 4:3) x batch(bits 2:0).
// Each lane keeps a 4x4 running product P (rows as float4).
// tri_up row data is streamed via async-to-LDS, double buffered per chunk.
// ---------------------------------------------------------------------------
__global__ __launch_bounds__(128) void amps_p1(
    const float* __restrict__ data, const float* __restrict__ tri,
    float* __restrict__ partials) {
  __shared__ __align__(16) char lds[4 * WAVE_LDS];
  const int col   = blockIdx.x;          // 0..1022
  const int len   = col + 1;             // chain length
  const int tid   = threadIdx.x;
  const int wavek = tid >> 5;
  const int lane  = tid & 31;
  if (wavek * (4 * SEG) >= len) return;  // whole wave inactive (uniform)
  const int slot   = lane >> 3;
  const int b      = lane & 7;
  const int seg    = wavek * 4 + slot;
  const int mstart = seg * SEG;
  const size_t rowblk = (size_t)col * (size_t)(col + 1) / 2;  // row start (blocks)
  char* ldsw = &lds[wavek * WAVE_LDS];
  const char* trib = (const char*)tri;

  // Issue chunk c (CHUNK blocks per slot, 4 slots, 8 x b128 wave-wide).
  auto issue = [&](int c) {
    const int buf = c & 1;
#pragma unroll
    for (int j = 0; j < 8; ++j) {
      const int sj = j & 3;              // slot serviced by this instruction
      const int pj = j >> 2;             // 512B half of the slot chunk
      const int ms = (wavek * 4 + sj) * SEG;
      int mb = ms + c * CHUNK + pj * 4 + (lane >> 3);   // block index in row
      if (mb > col) mb = col;            // clamp: stay inside this row
      const char* g = trib + (((rowblk + (size_t)mb) << 7) |
                              (size_t)((lane & 7) << 4));
      char* l = ldsw + buf * BUF_STRIDE + sj * SLOT_STRIDE +
                (pj << 9) + (lane << 4);
      async_b128(g, l);
    }
  };

  f4 Pr0 = {1.f,0.f,0.f,0.f}, Pr1 = {0.f,1.f,0.f,0.f},
     Pr2 = {0.f,0.f,1.f,0.f}, Pr3 = {0.f,0.f,0.f,1.f};
  const float* drow = data + b * NPOS;

  issue(0);
  for (int c = 0; c < NCHUNKS; ++c) {
    const int m0 = mstart + c * CHUNK;
    int ga = m0;     if (ga > NPOS - 4) ga = NPOS - 4;   // clamped groups of 4
    int gb = m0 + 4; if (gb > NPOS - 4) gb = NPOS - 4;   // (exact when active)
    const f4 da = *(const f4*)(drow + ga);
    const f4 db = *(const f4*)(drow + gb);
    if (c + 1 < NCHUNKS) { issue(c + 1); WAIT_ASYNC(8); }  // in-order async done
    else                 { WAIT_ASYNC(0); }
    const f4* q = (const f4*)(ldsw + (c & 1) * BUF_STRIDE + slot * SLOT_STRIDE);
    const float dv[8] = {da.x, da.y, da.z, da.w, db.x, db.y, db.z, db.w};
#pragma unroll
    for (int t = 0; t < CHUNK; ++t) {
      // block layout [l][r][p], p fastest -> q[k] = {T0,T1,T0,T1} for 2 (l,r)
      const f4 q0=q[t*8+0], q1=q[t*8+1], q2=q[t*8+2], q3=q[t*8+3],
               q4=q[t*8+4], q5=q[t*8+5], q6=q[t*8+6], q7=q[t*8+7];
      const float e0 = dv[t], e1 = 1.0f - e0;
      f4 A0 = { q0.x*e0+q0.y*e1, q0.z*e0+q0.w*e1, q1.x*e0+q1.y*e1, q1.z*e0+q1.w*e1 };
      f4 A1 = { q2.x*e0+q2.y*e1, q2.z*e0+q2.w*e1, q3.x*e0+q3.y*e1, q3.z*e0+q3.w*e1 };
      f4 A2 = { q4.x*e0+q4.y*e1, q4.z*e0+q4.w*e1, q5.x*e0+q5.y*e1, q5.z*e0+q5.w*e1 };
      f4 A3 = { q6.x*e0+q6.y*e1, q6.z*e0+q6.w*e1, q7.x*e0+q7.y*e1, q7.z*e0+q7.w*e1 };
      const bool act = (m0 + t) < len;           // tail steps multiply by I
      const f4 I0={1.f,0.f,0.f,0.f}, I1={0.f,1.f,0.f,0.f},
               I2={0.f,0.f,1.f,0.f}, I3={0.f,0.f,0.f,1.f};
      A0 = act ? A0 : I0;  A1 = act ? A1 : I1;
      A2 = act ? A2 : I2;  A3 = act ? A3 : I3;
      const f4 n0 = Pr0.x*A0 + Pr0.y*A1 + Pr0.z*A2 + Pr0.w*A3;  // P <- P @ A
      const f4 n1 = Pr1.x*A0 + Pr1.y*A1 + Pr1.z*A2 + Pr1.w*A3;
      const f4 n2 = Pr2.x*A0 + Pr2.y*A1 + Pr2.z*A2 + Pr2.w*A3;
      const f4 n3 = Pr3.x*A0 + Pr3.y*A1 + Pr3.z*A2 + Pr3.w*A3;
      Pr0 = n0; Pr1 = n1; Pr2 = n2; Pr3 = n3;
    }
  }
  if (mstart < len) {                            // store only active segments
    f4* dst = (f4*)(partials +
                    ((((size_t)col * NSEGMAX + (size_t)seg) * 8 + (size_t)b) << 4));
    dst[0] = Pr0; dst[1] = Pr1; dst[2] = Pr2; dst[3] = Pr3;
  }
}

// ---------------------------------------------------------------------------
// Phase 2: per (column, batch): v = e0^T * prod(partials), logits via diag,
// log-softmax contribution per position.  colid == NCOL handles n = 0.
// ---------------------------------------------------------------------------
__global__ __launch_bounds__(256) void amps_p2(
    const float* __restrict__ data, const float* __restrict__ diag,
    const float* __restrict__ partials, float* __restrict__ contrib) {
  const int id    = blockIdx.x * 256 + threadIdx.x;   // 0..8191
  const int colid = id >> 3;
  const int b     = id & 7;
  float x0, x1; int pos;
  if (colid < NCOL) {
    const int nseg = (colid + SEG) >> 6;              // ceil((colid+1)/64)
    f4 v = {1.f, 0.f, 0.f, 0.f};
    for (int j = 0; j < nseg; ++j) {
      const f4* Pm = (const f4*)(partials +
                     ((((size_t)colid * NSEGMAX + (size_t)j) * 8 + (size_t)b) << 4));
      const f4 r0 = Pm[0], r1 = Pm[1], r2 = Pm[2], r3 = Pm[3];
      v = v.x * r0 + v.y * r1 + v.z * r2 + v.w * r3;  // v <- v @ P_j
    }
    const float* dg = diag + (size_t)(colid + 1) * 8; // diag[n][d][p]
    x0 = v.x*dg[0] + v.y*dg[2] + v.z*dg[4] + v.w*dg[6];
    x1 = v.x*dg[1] + v.y*dg[3] + v.z*dg[5] + v.w*dg[7];
    pos = colid + 1;
  } else {                                            // n = 0: diag[0,0,:]
    x0 = diag[0]; x1 = diag[1]; pos = 0;
  }
  const float dv  = data[b * NPOS + pos];
  const float mx  = fmaxf(x0, x1);
  const float lse = mx + logf(expf(x0 - mx) + expf(x1 - mx));
  contrib[b * NPOS + pos] = x0 * dv + x1 * (1.0f - dv) - lse;
}

// ---------------------------------------------------------------------------
// Phase 3: reduce contrib[8][1024] -> out[8].  Per-wave shuffle reduce to 4
// partials per batch, then one v_wmma_f32_16x16x4_f32 (A = 8x4 partials,
// B = ones) finishes the sum on the matrix pipe; lane 0 holds out[b] in D[b].
// ---------------------------------------------------------------------------
__global__ __launch_bounds__(256) void amps_p3(
    const float* __restrict__ contrib, float* __restrict__ out) {
  __shared__ float sh[32];
  const int tid = threadIdx.x;
  const int w = tid >> 5, lane = tid & 31;    // wave w <-> batch w
  float v = 0.0f;
  const float* row = contrib + w * NPOS;
  for (int i = lane; i < NPOS; i += 32) v += row[i];
  v += __shfl_xor(v, 16, 32);
  v += __shfl_xor(v, 8, 32);
  v += __shfl_xor(v, 4, 32);                  // lanes 0..3: 4 partials/batch
  if (lane < 4) sh[w * 4 + lane] = v;
  __syncthreads();
  if (tid < 32) {                             // whole wave 0, EXEC all-1s
    const int aM = lane & 15;                 // A-matrix row (M)
    const int kh = lane >> 4;                 // 0 -> K={0,1}, 1 -> K={2,3}
    v2f a;
    a.x = (aM < 8) ? sh[aM * 4 + kh * 2 + 0] : 0.0f;
    a.y = (aM < 8) ? sh[aM * 4 + kh * 2 + 1] : 0.0f;
    v2f bb; bb.x = 1.0f; bb.y = 1.0f;         // B = ones(4x16)
    v8f c = {};
    v8f d = __builtin_amdgcn_wmma_f32_16x16x4_f32(
        false, a, false, bb, (short)0, c, false, false);
    if (lane == 0) {                          // lane 0: D[v] = (M=v, N=0)
      out[0] = d[0]; out[1] = d[1]; out[2] = d[2]; out[3] = d[3];
      out[4] = d[4]; out[5] = d[5]; out[6] = d[6]; out[7] = d[7];
    }
  }
}

// ---------------------------------------------------------------------------
extern "C" void kernel_launch(void* const* d_in, const int* in_sizes, int n_in,
                              void* d_out, int out_size, void* d_ws, size_t ws_size,
                              hipStream_t stream) {
  const float* data = (const float*)d_in[0];   // (8, 1024)
  const float* tri  = (const float*)d_in[1];   // (523776, 4, 4, 2)
  const float* diag = (const float*)d_in[2];   // (1024, 4, 2)
  float* partials = (float*)d_ws;              // 1023*16*8*16 floats (~8.4 MB)
  float* contrib  = partials + (size_t)NCOL * NSEGMAX * 8 * 16;  // 8192 floats
  (void)in_sizes; (void)n_in; (void)out_size; (void)ws_size;

  amps_p1<<<dim3(NCOL), dim3(128), 0, stream>>>(data, tri, partials);
  amps_p2<<<dim3(32),   dim3(256), 0, stream>>>(data, diag, partials, contrib);
  amps_p3<<<dim3(1),    dim3(256), 0, stream>>>(contrib, (float*)d_out);
}